// GATv3_23862838296802
// MI455X (gfx1250) — compile-verified
//
#include <hip/hip_runtime.h>
#include <math.h>

typedef float v2f __attribute__((ext_vector_type(2)));
typedef float v8f __attribute__((ext_vector_type(8)));

#define BN_EPS 1e-5f

__device__ __forceinline__ float lrelu01(float v) { return v >= 0.f ? v : 0.01f * v; }

// ---------------------------------------------------------------------------
// WMMA fp32 GEMM: out[M,Nout] = act(A[M,K] @ W[K,Nout] + bias)
// One wave computes one 16x16 tile of C via V_WMMA_F32_16X16X4_F32, K step 4.
// Requires M, Nout multiples of 16, K multiple of 4 (true for all calls here).
// ---------------------------------------------------------------------------
__global__ __launch_bounds__(128) void wmma_gemm_f32(
    const float* __restrict__ A, const float* __restrict__ W,
    const float* __restrict__ bias, float* __restrict__ out,
    int M, int K, int Nout, int act) {
  const int lane = threadIdx.x & 31;
  const int wave = threadIdx.x >> 5;
  const int Mtiles = M >> 4;
  const int mt = blockIdx.x * (blockDim.x >> 5) + wave;
  if (mt >= Mtiles) return;                 // wave-uniform guard
  const int nt = blockIdx.y;
  const int m0 = mt << 4;
  const int n0 = nt << 4;
  const int r  = lane & 15;                 // A row / B col within tile
  const int kb = (lane >> 4) << 1;          // K sub-offset: lanes 0-15 -> K=0,1; 16-31 -> K=2,3

  v8f acc = {};
  const float* Arow = A + (size_t)(m0 + r) * K + kb;
  const float* Bcol = W + (size_t)kb * Nout + (n0 + r);
  for (int k = 0; k < K; k += 4) {
    v2f a, b;
    a.x = Arow[k];
    a.y = Arow[k + 1];
    b.x = Bcol[(size_t)k * Nout];
    b.y = Bcol[(size_t)(k + 1) * Nout];
    acc = __builtin_amdgcn_wmma_f32_16x16x4_f32(
        /*neg_a=*/false, a, /*neg_b=*/false, b,
        /*c_mod=*/(short)0, acc, /*reuse_a=*/false, /*reuse_b=*/false);
  }

  // C/D layout: VGPR v, lanes 0-15 -> M=v, N=lane; lanes 16-31 -> M=v+8.
  const int half = lane >> 4;
  const int n = n0 + r;
  const float bval = bias ? bias[n] : 0.0f;
#pragma unroll
  for (int v = 0; v < 8; ++v) {
    int m = m0 + v + half * 8;
    float val = acc[v] + bval;
    if (act) val = lrelu01(val);
    out[(size_t)m * Nout + n] = val;
  }
}

// ---------------------------------------------------------------------------
// Utility fills
// ---------------------------------------------------------------------------
__global__ void fill_f32(float* p, float v, size_t n) {
  size_t i = (size_t)blockIdx.x * blockDim.x + threadIdx.x;
  if (i < n) p[i] = v;
}

__global__ void fill_bias_rows(float* p, const float* __restrict__ b, int H, size_t total) {
  size_t i = (size_t)blockIdx.x * blockDim.x + threadIdx.x;
  if (i < total) p[i] = b[(int)(i % (size_t)H)];
}

// ---------------------------------------------------------------------------
// Attention source/dest scores: ss[n] = <hW[n], a_s>, sd[n] = <hW[n], a_d>
// One wave32 per node.
// ---------------------------------------------------------------------------
__global__ void score_kernel(const float* __restrict__ hW,
                             const float* __restrict__ as_, const float* __restrict__ ad_,
                             float* __restrict__ ss, float* __restrict__ sd, int N, int H) {
  int node = (int)((blockIdx.x * (size_t)blockDim.x + threadIdx.x) >> 5);
  int lane = threadIdx.x & 31;
  if (node >= N) return;
  const float* r = hW + (size_t)node * H;
  float s0 = 0.f, s1 = 0.f;
  for (int j = lane; j < H; j += 32) {
    float v = r[j];
    s0 += v * as_[j];
    s1 += v * ad_[j];
  }
  for (int o = 16; o; o >>= 1) {
    s0 += __shfl_xor(s0, o);
    s1 += __shfl_xor(s1, o);
  }
  if (lane == 0) { ss[node] = s0; sd[node] = s1; }
}

__device__ __forceinline__ void atomicMaxFloat(float* addr, float val) {
  if (val >= 0.f) atomicMax((int*)addr, __float_as_int(val));
  else            atomicMin((unsigned int*)addr, __float_as_uint(val));
}

__device__ __forceinline__ void edge_sd(const int* __restrict__ ei, int E, int i, int& s, int& d) {
  if (i < E) { s = ei[i]; d = ei[E + i]; }
  else       { s = d = i - E; }
}

// e[i] = LeakyReLU(ss[src]+sd[dst], 0.2); segment max into m[dst]
__global__ void edge_score_max(const int* __restrict__ ei, int E, int Ntot,
                               const float* __restrict__ ss, const float* __restrict__ sd,
                               float* __restrict__ e, float* __restrict__ m) {
  int i = blockIdx.x * blockDim.x + threadIdx.x;
  if (i >= Ntot) return;
  int s, d; edge_sd(ei, E, i, s, d);
  float v = ss[s] + sd[d];
  v = v >= 0.f ? v : 0.2f * v;
  e[i] = v;
  atomicMaxFloat(m + d, v);
}

// e[i] = exp(e[i]-m[dst]); segment sum into denom[dst]
__global__ void edge_exp_sum(const int* __restrict__ ei, int E, int Ntot,
                             const float* __restrict__ m, float* __restrict__ e,
                             float* __restrict__ denom) {
  int i = blockIdx.x * blockDim.x + threadIdx.x;
  if (i >= Ntot) return;
  int s, d; edge_sd(ei, E, i, s, d);
  float ex = expf(e[i] - m[d]);
  e[i] = ex;
  atomicAdd(denom + d, ex);
}

// e[i] /= denom[dst]  (alpha)
__global__ void edge_alpha(const int* __restrict__ ei, int E, int Ntot,
                           const float* __restrict__ denom, float* __restrict__ e) {
  int i = blockIdx.x * blockDim.x + threadIdx.x;
  if (i >= Ntot) return;
  int s, d; edge_sd(ei, E, i, s, d);
  e[i] /= denom[d];
}

// agg[dst,:] += alpha[i] * hW[src,:]  — one thread per (edge, channel)
__global__ void edge_aggregate(const int* __restrict__ ei, int E, int Ntot,
                               const float* __restrict__ alpha, const float* __restrict__ hW,
                               float* __restrict__ agg, int H) {
  size_t t = (size_t)blockIdx.x * blockDim.x + threadIdx.x;
  size_t total = (size_t)Ntot * H;
  if (t >= total) return;
  int i = (int)(t / (size_t)H);
  int j = (int)(t % (size_t)H);
  int s, d; edge_sd(ei, E, i, s, d);
  atomicAdd(agg + (size_t)d * H + j, alpha[i] * hW[(size_t)s * H + j]);
}

// ---------------------------------------------------------------------------
// BatchNorm over the node dimension: one block per channel
// ---------------------------------------------------------------------------
__global__ __launch_bounds__(256) void bn_stats(const float* __restrict__ x, int N, int H,
                                                float* __restrict__ mu, float* __restrict__ var) {
  __shared__ float s1[256], s2[256];
  int c = blockIdx.x;
  float a = 0.f, b = 0.f;
  for (int n = threadIdx.x; n < N; n += blockDim.x) {
    float v = x[(size_t)n * H + c];
    a += v; b += v * v;
  }
  s1[threadIdx.x] = a; s2[threadIdx.x] = b;
  __syncthreads();
  for (int o = 128; o; o >>= 1) {
    if ((int)threadIdx.x < o) { s1[threadIdx.x] += s1[threadIdx.x + o]; s2[threadIdx.x] += s2[threadIdx.x + o]; }
    __syncthreads();
  }
  if (threadIdx.x == 0) {
    float mean = s1[0] / (float)N;
    mu[c] = mean;
    var[c] = s2[0] / (float)N - mean * mean;
  }
}

// out = lrelu( g*(x-mu)/sqrt(var+eps)+b ) + identity
__global__ void bn_apply_res(const float* __restrict__ x,
                             const float* __restrict__ mu, const float* __restrict__ var,
                             const float* __restrict__ g, const float* __restrict__ b,
                             const float* __restrict__ identity, float* __restrict__ out,
                             size_t total, int H) {
  size_t i = (size_t)blockIdx.x * blockDim.x + threadIdx.x;
  if (i >= total) return;
  int c = (int)(i % (size_t)H);
  float v = g[c] * (x[i] - mu[c]) * rsqrtf(var[c] + BN_EPS) + b[c];
  out[i] = lrelu01(v) + identity[i];
}

// ---------------------------------------------------------------------------
// Poincare expmap0 + segment mean-pool (one wave32 per node)
// ---------------------------------------------------------------------------
__global__ void pool_kernel(const float* __restrict__ h, const int* __restrict__ batch,
                            float* __restrict__ pooled, float* __restrict__ cnt, int N, int H) {
  int node = (int)((blockIdx.x * (size_t)blockDim.x + threadIdx.x) >> 5);
  int lane = threadIdx.x & 31;
  if (node >= N) return;
  const float* r = h + (size_t)node * H;
  float sq = 0.f;
  for (int j = lane; j < H; j += 32) { float v = r[j]; sq += v * v; }
  for (int o = 16; o; o >>= 1) sq += __shfl_xor(sq, o);
  float nrm = sqrtf(sq);
  if (nrm < 1e-15f) nrm = 1e-15f;
  float scale = tanhf(nrm) / nrm;
  int g = batch[node];
  for (int j = lane; j < H; j += 32)
    atomicAdd(pooled + (size_t)g * H + j, r[j] * scale);
  if (lane == 0) atomicAdd(cnt + g, 1.0f);
}

__global__ void pool_div(float* pooled, const float* __restrict__ cnt, int G, int H) {
  int i = blockIdx.x * blockDim.x + threadIdx.x;
  if (i >= G * H) return;
  float c = cnt[i / H];
  if (c < 1.f) c = 1.f;
  pooled[i] /= c;
}

// small dense layer (used for fc4: [G,48]@[48,4])
__global__ void fc_small(const float* __restrict__ A, const float* __restrict__ W,
                         const float* __restrict__ b, float* __restrict__ out,
                         int M, int K, int Nout) {
  int i = blockIdx.x * blockDim.x + threadIdx.x;
  if (i >= M * Nout) return;
  int m = i / Nout, n = i % Nout;
  float s = b[n];
  for (int k = 0; k < K; ++k) s += A[m * K + k] * W[k * Nout + n];
  out[i] = s;
}

// ---------------------------------------------------------------------------
extern "C" void kernel_launch(void* const* d_in, const int* in_sizes, int n_in,
                              void* d_out, int out_size, void* d_ws, size_t ws_size,
                              hipStream_t stream) {
  const float* x       = (const float*)d_in[0];
  const int*   ei      = (const int*)d_in[1];
  const int*   batch   = (const int*)d_in[2];
  const float* embed_W = (const float*)d_in[3];
  const float* embed_b = (const float*)d_in[4];
  const float* convW[3]  = {(const float*)d_in[5],  (const float*)d_in[9],  (const float*)d_in[13]};
  const float* convAs[3] = {(const float*)d_in[6],  (const float*)d_in[10], (const float*)d_in[14]};
  const float* convAd[3] = {(const float*)d_in[7],  (const float*)d_in[11], (const float*)d_in[15]};
  const float* convB[3]  = {(const float*)d_in[8],  (const float*)d_in[12], (const float*)d_in[16]};
  const float* fc1_W = (const float*)d_in[17]; const float* fc1_b = (const float*)d_in[18];
  const float* fc2_W = (const float*)d_in[19]; const float* fc2_b = (const float*)d_in[20];
  const float* bng[3] = {(const float*)d_in[21], (const float*)d_in[23], (const float*)d_in[25]};
  const float* bnb[3] = {(const float*)d_in[22], (const float*)d_in[24], (const float*)d_in[26]};
  const float* fc3_W = (const float*)d_in[27]; const float* fc3_b = (const float*)d_in[28];
  const float* fc4_W = (const float*)d_in[29]; const float* fc4_b = (const float*)d_in[30];

  const int H    = in_sizes[6];            // 96
  const int F    = in_sizes[3] / H;        // 256
  const int N    = in_sizes[0] / F;        // 50000
  const int E    = in_sizes[1] / 2;        // 800000
  const int H2   = in_sizes[28];           // 48
  const int NC   = in_sizes[n_in - 1];     // 4
  const int G    = out_size / NC;          // 64
  const int Ntot = E + N;                  // edges incl. self-loops
  const size_t NH = (size_t)N * H;

  // ---- workspace carve ----
  char* wp = (char*)d_ws;
  auto carve = [&](size_t nf) -> float* {
    float* p = (float*)wp;
    wp += ((nf * sizeof(float) + 255) / 256) * 256;
    return p;
  };
  float* identity = carve(NH);   // embed output, residual source
  float* big1     = carve(NH);
  float* big2     = carve(NH);
  float* agg      = carve(NH);
  float* ss       = carve(N);
  float* sd       = carve(N);
  float* mmax     = carve(N);
  float* denom    = carve(N);
  float* ebuf     = carve((size_t)Ntot);
  float* mu       = carve(H);
  float* var      = carve(H);
  float* pooled   = carve((size_t)G * H);
  float* pooled2  = carve((size_t)G * H2);
  float* cnt      = carve(G);
  (void)ws_size;

  auto cdiv = [](size_t a, size_t b) -> unsigned { return (unsigned)((a + b - 1) / b); };

  auto gemm = [&](const float* A, const float* W, const float* bias, float* out,
                  int M, int K, int Nout, int act) {
    int Mtiles = M / 16;
    dim3 grid((Mtiles + 3) / 4, Nout / 16);
    wmma_gemm_f32<<<grid, 128, 0, stream>>>(A, W, bias, out, M, K, Nout, act);
  };

  // GATConv + BN + lrelu + residual: hout = lrelu(bn(gat(hin))) + identity
  auto run_conv = [&](int li, const float* hin, float* hW, float* hout) {
    gemm(hin, convW[li], nullptr, hW, N, H, H, 0);
    score_kernel<<<cdiv((size_t)N * 32, 256), 256, 0, stream>>>(hW, convAs[li], convAd[li], ss, sd, N, H);
    fill_f32<<<cdiv(N, 256), 256, 0, stream>>>(mmax, -INFINITY, (size_t)N);
    fill_f32<<<cdiv(N, 256), 256, 0, stream>>>(denom, 0.f, (size_t)N);
    edge_score_max<<<cdiv(Ntot, 256), 256, 0, stream>>>(ei, E, Ntot, ss, sd, ebuf, mmax);
    edge_exp_sum<<<cdiv(Ntot, 256), 256, 0, stream>>>(ei, E, Ntot, mmax, ebuf, denom);
    edge_alpha<<<cdiv(Ntot, 256), 256, 0, stream>>>(ei, E, Ntot, denom, ebuf);
    fill_bias_rows<<<cdiv(NH, 256), 256, 0, stream>>>(agg, convB[li], H, NH);
    edge_aggregate<<<cdiv((size_t)Ntot * H, 256), 256, 0, stream>>>(ei, E, Ntot, ebuf, hW, agg, H);
    bn_stats<<<H, 256, 0, stream>>>(agg, N, H, mu, var);
    bn_apply_res<<<cdiv(NH, 256), 256, 0, stream>>>(agg, mu, var, bng[li], bnb[li], identity, hout, NH, H);
  };

  // 1) embed: identity = lrelu(x @ embed_W + b)
  gemm(x, embed_W, embed_b, identity, N, F, H, 1);

  // 2) conv1 (in: identity) -> big2 ; fc1 -> big1
  run_conv(0, identity, /*hW=*/big1, /*hout=*/big2);
  gemm(big2, fc1_W, fc1_b, big1, N, H, H, 1);

  // 3) conv2 (in: big1) -> big1 (input consumed once hW built) ; fc2 -> big2
  run_conv(1, big1, /*hW=*/big2, /*hout=*/big1);
  gemm(big1, fc2_W, fc2_b, big2, N, H, H, 1);

  // 4) conv3 (in: big2) -> big2
  run_conv(2, big2, /*hW=*/big1, /*hout=*/big2);

  // 5) Poincare expmap0 + mean pool per graph
  fill_f32<<<cdiv((size_t)G * H, 256), 256, 0, stream>>>(pooled, 0.f, (size_t)G * H);
  fill_f32<<<cdiv(G, 256), 256, 0, stream>>>(cnt, 0.f, (size_t)G);
  pool_kernel<<<cdiv((size_t)N * 32, 256), 256, 0, stream>>>(big2, batch, pooled, cnt, N, H);
  pool_div<<<cdiv((size_t)G * H, 256), 256, 0, stream>>>(pooled, cnt, G, H);

  // 6) fc3 (WMMA, lrelu) + fc4 (tiny)
  gemm(pooled, fc3_W, fc3_b, pooled2, G, H, H2, 1);
  fc_small<<<cdiv((size_t)G * NC, 256), 256, 0, stream>>>(pooled2, fc4_W, fc4_b, (float*)d_out, G, H2, NC);
}